// MMSolver_1640677507627
// MI455X (gfx1250) — compile-verified
//
#include <hip/hip_runtime.h>

// ---------------- problem constants (from reference) ----------------
#define NXg 256
#define NYg 256
#define TSTEPSc 256
#define NSIGc 2
#define NSRCc 3
#define NPRBc 5
#define RELAXNc 100

#define ALPHA_RUNc 0.01f
#define ALPHA_RELAXc 0.5f
#define CSOTc 1.0e-4f

// ---------------- decomposition ----------------
#define NBLK 32     // persistent workgroups, one per WGP (216KB LDS each)
#define ROWSB 8     // interior rows per block
#define HALO 4      // halo depth for fully fused 4-stage RK4 (one global sync/step)
#define NRT (ROWSB + 2*HALO)  // 16 tile rows
#define NTX 256     // one lane per column
#define NTY 4       // row-slices -> 1024 threads = 32 waves = 8 waves/SIMD32
#define IDX3(c,r,yy) ((((c)*NRT)+(r))*NYg+(yy))
#define GIDX(c,r,yy) ((((c)*NXg)+(r))*NYg+(yy))
#define AIDX(c,r,yy) ((((c)*ROWSB)+(r))*NYg+(yy))

struct Ctx {
  float *Bex, *M, *SA, *SB, *ACC;
  float cex, cdem, h;
  int r0;
};

__device__ __forceinline__ void wait_async0()
{
#if __has_builtin(__builtin_amdgcn_s_wait_asynccnt)
  __builtin_amdgcn_s_wait_asynccnt(0);
#else
  asm volatile("s_wait_asynccnt 0" ::: "memory");
#endif
}

// Device-wide barrier: monotonic counter (reset by memset each launch) +
// CDNA5 cluster barrier (NOP when not cluster-dispatched, per CDNA5 ISA).
__device__ __forceinline__ void gsync(unsigned* cnt, unsigned& gen)
{
  __threadfence();
  __syncthreads();
  if (threadIdx.x == 0 && threadIdx.y == 0) {
    const unsigned g = ++gen;
    __hip_atomic_fetch_add(cnt, 1u, __ATOMIC_RELEASE, __HIP_MEMORY_SCOPE_AGENT);
    while (__hip_atomic_load(cnt, __ATOMIC_ACQUIRE, __HIP_MEMORY_SCOPE_AGENT) <
           g * (unsigned)NBLK)
      __builtin_amdgcn_s_sleep(2);
  }
  __builtin_amdgcn_s_cluster_barrier();  // s_barrier_signal/-wait -3 (gfx1250)
  __syncthreads();
  __threadfence();
}

// Tile load via CDNA5 async global->LDS DMA (GLOBAL_LOAD_ASYNC_TO_LDS_B128,
// ASYNCcnt). Per-lane addresses carry the clamped ('edge' padded) source row.
// 1024 lanes x 16B = 16KB (one 16-row component plane) per issue; 3 issues total.
__device__ __forceinline__ void load_tile(const Ctx& cx, const float* __restrict__ g_m)
{
  const int tid = (int)(threadIdx.y * NTX + threadIdx.x);
  const int lr  = tid >> 6;            // 0..15 tile row
  const int cb  = (tid & 63) * 16;     // byte offset within the 1KB row
  int gr = cx.r0 - HALO + lr;
  gr = gr < 0 ? 0 : (gr > NXg - 1 ? NXg - 1 : gr);   // 'edge' padding
  #pragma unroll
  for (int c = 0; c < 3; ++c) {
    // Low 32 bits of a generic LDS pointer are the workgroup LDS byte offset.
    const unsigned ldsOff =
        (unsigned)(size_t)(const void*)(&cx.M[IDX3(c, lr, 0)]) + (unsigned)cb;
    const unsigned long long gaddr =
        (unsigned long long)(size_t)(const void*)(&g_m[GIDX(c, gr, 0)]) +
        (unsigned long long)cb;
    asm volatile("global_load_async_to_lds_b128 %0, %1, off"
                 :: "v"(ldsOff), "v"(gaddr) : "memory");
  }
  wait_async0();
  __syncthreads();
}

// One RK4 stage: torque on the shrinking-halo region (rows strided over ty),
// write next stage field, accumulate weighted k on the interior.
__device__ __forceinline__ void stage_eval(const Ctx& cx, const float* __restrict__ Sin,
                                           float* __restrict__ Sout,
                                           int s, float cstep, float w,
                                           float alpha, float inva)
{
  const int y  = (int)threadIdx.x;
  const int ty = (int)threadIdx.y;
  const int yl = (y == 0) ? 0 : y - 1;
  const int yr = (y == NYg - 1) ? (NYg - 1) : y + 1;
  for (int lr = 1 + s + ty; lr <= NRT - 2 - s; lr += NTY) {
    const int gr = cx.r0 - HALO + lr;
    const int ru = (gr <= 0) ? lr : lr - 1;          // clamp at global edges
    const int rd = (gr >= NXg - 1) ? lr : lr + 1;
    const float mx = Sin[IDX3(0, lr, y)];
    const float my = Sin[IDX3(1, lr, y)];
    const float mz = Sin[IDX3(2, lr, y)];
    const float lapx = Sin[IDX3(0,ru,y)] + Sin[IDX3(0,rd,y)] + Sin[IDX3(0,lr,yl)] + Sin[IDX3(0,lr,yr)] - 4.0f*mx;
    const float lapy = Sin[IDX3(1,ru,y)] + Sin[IDX3(1,rd,y)] + Sin[IDX3(1,lr,yl)] + Sin[IDX3(1,lr,yr)] - 4.0f*my;
    const float lapz = Sin[IDX3(2,ru,y)] + Sin[IDX3(2,rd,y)] + Sin[IDX3(2,lr,yl)] + Sin[IDX3(2,lr,yr)] - 4.0f*mz;
    const float Bx = cx.Bex[IDX3(0, lr, y)] + cx.cex * lapx;
    const float By = cx.Bex[IDX3(1, lr, y)] + cx.cex * lapy;
    const float Bz = cx.Bex[IDX3(2, lr, y)] + cx.cex * lapz - cx.cdem * mz;  // demag (0,0,1)
    // t = m x Beff
    const float tx = my * Bz - mz * By;
    const float ty_ = mz * Bx - mx * Bz;
    const float tz = mx * By - my * Bx;
    // u = m x t
    const float ux = my * tz - mz * ty_;
    const float uy = mz * tx - mx * tz;
    const float uz = mx * ty_ - my * tx;
    // SOT with p=(0,1,0): m x (m x p) = (mx*my, -(mz^2+mx^2), my*mz)
    const float kx = -inva * (tx  + alpha * ux) + CSOTc * (mx * my);
    const float ky = -inva * (ty_ + alpha * uy) - CSOTc * (mz * mz + mx * mx);
    const float kz = -inva * (tz  + alpha * uz) + CSOTc * (my * mz);
    if (Sout) {
      Sout[IDX3(0, lr, y)] = cx.M[IDX3(0, lr, y)] + cstep * kx;
      Sout[IDX3(1, lr, y)] = cx.M[IDX3(1, lr, y)] + cstep * ky;
      Sout[IDX3(2, lr, y)] = cx.M[IDX3(2, lr, y)] + cstep * kz;
    }
    if (lr >= HALO && lr < HALO + ROWSB) {
      const int ar = lr - HALO;
      if (s == 0) {
        cx.ACC[AIDX(0, ar, y)] = kx;
        cx.ACC[AIDX(1, ar, y)] = ky;
        cx.ACC[AIDX(2, ar, y)] = kz;
      } else {
        cx.ACC[AIDX(0, ar, y)] += w * kx;
        cx.ACC[AIDX(1, ar, y)] += w * ky;
        cx.ACC[AIDX(2, ar, y)] += w * kz;
      }
    }
  }
}

__device__ __forceinline__ void rk4_stages(const Ctx& cx, float alpha)
{
  const float inva = 1.0f / (1.0f + alpha * alpha);
  stage_eval(cx, cx.M,  cx.SA,   0, 0.5f * cx.h, 1.0f, alpha, inva); __syncthreads();
  stage_eval(cx, cx.SA, cx.SB,   1, 0.5f * cx.h, 2.0f, alpha, inva); __syncthreads();
  stage_eval(cx, cx.SB, cx.SA,   2, cx.h,        2.0f, alpha, inva); __syncthreads();
  stage_eval(cx, cx.SA, nullptr, 3, 0.0f,        1.0f, alpha, inva); __syncthreads();
}

__device__ __forceinline__ void writeback(const Ctx& cx, float* __restrict__ g_m)
{
  const int y  = (int)threadIdx.x;
  const float h6 = cx.h * (1.0f / 6.0f);
  for (int ar = (int)threadIdx.y; ar < ROWSB; ar += NTY) {
    const int lr = ar + HALO;
    const int gr = cx.r0 + ar;
    g_m[GIDX(0, gr, y)] = cx.M[IDX3(0, lr, y)] + h6 * cx.ACC[AIDX(0, ar, y)];
    g_m[GIDX(1, gr, y)] = cx.M[IDX3(1, lr, y)] + h6 * cx.ACC[AIDX(1, ar, y)];
    g_m[GIDX(2, gr, y)] = cx.M[IDX3(2, lr, y)] + h6 * cx.ACC[AIDX(2, ar, y)];
  }
}

__global__ __launch_bounds__(NTX*NTY, 1)
void mm_kernel(float* __restrict__ g_m, float* __restrict__ g_mrel,
               unsigned* __restrict__ g_bar,
               const float* __restrict__ sig, const float* __restrict__ g_Bext,
               const float* __restrict__ g_Ms, const int* __restrict__ srcp,
               const int* __restrict__ prbp, const int* __restrict__ fbp,
               float* __restrict__ out)
{
  extern __shared__ float lds[];
  const int y  = (int)threadIdx.x;
  const int ty = (int)threadIdx.y;
  Ctx cx;
  cx.Bex = lds;
  cx.M   = cx.Bex + 3 * NRT * NYg;
  cx.SA  = cx.M   + 3 * NRT * NYg;
  cx.SB  = cx.SA  + 3 * NRT * NYg;
  cx.ACC = cx.SB  + 3 * NRT * NYg;
  cx.r0  = (int)blockIdx.x * ROWSB;

  const float Ms = g_Ms[0];
  cx.h    = 175950000000.0f * 5.0e-12f;                      // GAMMA_LL * DT
  cx.cex  = 2.0f * 3.5e-12f / (Ms * 5.0e-8f * 5.0e-8f);      // exchange coeff
  cx.cdem = (4.0e-7f * 3.14159265358979323846f) * Ms;        // MU0 * Msat
  unsigned gen = 0;

  // Cache B_ext tile in LDS (constant over the whole run)
  for (int c = 0; c < 3; ++c)
    for (int lr = ty; lr < NRT; lr += NTY) {
      int gr = cx.r0 - HALO + lr;
      gr = gr < 0 ? 0 : (gr > NXg - 1 ? NXg - 1 : gr);
      cx.Bex[IDX3(c, lr, y)] = g_Bext[GIDX(c, gr, y)];
    }
  // m0 = (0,1,0) on own rows
  for (int ar = ty; ar < ROWSB; ar += NTY) {
    const int gr = cx.r0 + ar;
    g_m[GIDX(0, gr, y)] = 0.0f;
    g_m[GIDX(1, gr, y)] = 1.0f;
    g_m[GIDX(2, gr, y)] = 0.0f;
  }
  gsync(g_bar, gen);

  // ---- relaxation: 100 steps, alpha = 0.5, no sources ----
  for (int it = 0; it < RELAXNc; ++it) {
    load_tile(cx, g_m);
    rk4_stages(cx, ALPHA_RELAXc);
    writeback(cx, g_m);
    gsync(g_bar, gen);
  }
  // save relaxed state (own rows; same thread later reads them back)
  for (int ar = ty; ar < ROWSB; ar += NTY) {
    const int gr = cx.r0 + ar;
    g_mrel[GIDX(0, gr, y)] = g_m[GIDX(0, gr, y)];
    g_mrel[GIDX(1, gr, y)] = g_m[GIDX(1, gr, y)];
    g_mrel[GIDX(2, gr, y)] = g_m[GIDX(2, gr, y)];
  }

  const int finalb = fbp[0];
  int sr[NSRCc], scn[NSRCc], pr[NPRBc], pcn[NPRBc];
  for (int j = 0; j < NSRCc; ++j) { sr[j] = srcp[2 * j]; scn[j] = srcp[2 * j + 1]; }
  for (int p = 0; p < NPRBc; ++p) { pr[p] = prbp[2 * p]; pcn[p] = prbp[2 * p + 1]; }
  const float h6 = cx.h * (1.0f / 6.0f);

  // ---- driven runs: 2 signals x 256 steps, alpha = 0.01, sources + probes ----
  for (int sgn = 0; sgn < NSIGc; ++sgn) {
    for (int ar = ty; ar < ROWSB; ar += NTY) {   // reset to relaxed state (own rows)
      const int gr = cx.r0 + ar;
      g_m[GIDX(0, gr, y)] = g_mrel[GIDX(0, gr, y)];
      g_m[GIDX(1, gr, y)] = g_mrel[GIDX(1, gr, y)];
      g_m[GIDX(2, gr, y)] = g_mrel[GIDX(2, gr, y)];
    }
    gsync(g_bar, gen);
    for (int t = 0; t < TSTEPSc; ++t) {
      // patch LDS B_ext z with this step's source injection
      float saveB = 0.0f; int plr = -1, pcol = 0;
      if (ty == 0 && y < NSRCc) {
        const int lr = sr[y] - (cx.r0 - HALO);
        if (lr >= 0 && lr < NRT) {
          plr = lr; pcol = scn[y];
          saveB = cx.Bex[IDX3(2, plr, pcol)];
          cx.Bex[IDX3(2, plr, pcol)] = saveB + sig[(sgn * TSTEPSc + t) * NSRCc + y];
        }
      }
      __syncthreads();
      __builtin_prefetch((const void*)&g_m[GIDX(0, cx.r0, 0)], 0, 3);  // global_prefetch_b8
      load_tile(cx, g_m);
      rk4_stages(cx, ALPHA_RUNc);
      if (plr >= 0) cx.Bex[IDX3(2, plr, pcol)] = saveB;   // exact restore
      writeback(cx, g_m);
      // probes: recompute m_new z from LDS (M, ACC final after stage-3 barrier)
      if (ty == 0) {
        for (int p = 0; p < NPRBc; ++p) {
          const int gr = pr[p];
          if (gr >= cx.r0 && gr < cx.r0 + ROWSB && y == pcn[p]) {
            const int ar = gr - cx.r0;
            const float nz = cx.M[IDX3(2, ar + HALO, y)] + h6 * cx.ACC[AIDX(2, ar, y)];
            out[(sgn * TSTEPSc + t) * NPRBc + p] =
                finalb ? (nz - g_mrel[GIDX(2, gr, y)]) * Ms : nz;
          }
        }
      }
      gsync(g_bar, gen);
    }
  }
}

extern "C" void kernel_launch(void* const* d_in, const int* in_sizes, int n_in,
                              void* d_out, int out_size, void* d_ws, size_t ws_size,
                              hipStream_t stream)
{
  (void)in_sizes; (void)n_in; (void)out_size; (void)ws_size;
  const float* sig   = (const float*)d_in[0];   // (2,256,3)
  const float* Bext  = (const float*)d_in[1];   // (1,3,256,256)
  const float* Ms    = (const float*)d_in[2];   // scalar
  const int*   src   = (const int*)d_in[3];     // (3,2)
  const int*   probe = (const int*)d_in[4];     // (5,2)
  const int*   fb    = (const int*)d_in[5];     // scalar
  float* out = (float*)d_out;                   // (2,256,5)

  char* ws = (char*)d_ws;
  float*    g_m    = (float*)ws;                                        // 768 KB
  float*    g_mrel = (float*)(ws + (size_t)3 * NXg * NYg * 4);          // 768 KB
  unsigned* g_bar  = (unsigned*)(ws + (size_t)2 * 3 * NXg * NYg * 4);   // barrier

  hipMemsetAsync(g_bar, 0, 256, stream);  // reset barrier counter each launch

  const size_t ldsBytes = (size_t)(4 * 3 * NRT * NYg + 3 * ROWSB * NYg) * sizeof(float); // 216 KB
  hipFuncSetAttribute((const void*)mm_kernel,
                      hipFuncAttributeMaxDynamicSharedMemorySize, (int)ldsBytes);

  dim3 blk(NTX, NTY, 1);
  mm_kernel<<<NBLK, blk, ldsBytes, stream>>>(g_m, g_mrel, g_bar, sig, Bext, Ms,
                                             src, probe, fb, out);
}